// FPModule_13348758356091
// MI455X (gfx1250) — compile-verified
//
#include <hip/hip_runtime.h>

typedef __attribute__((ext_vector_type(2))) float v2f;
typedef __attribute__((ext_vector_type(8))) float v8f;

constexpr int N_PTS  = 4096;
constexpr int M_PTS  = 16384;
constexpr int C_IN   = 256;
constexpr int C_SKIP = 128;
constexpr int C_CAT  = C_IN + C_SKIP;   // 384
constexpr int HDIM   = 256;

// ---------------------------------------------------------------------------
// Kernel 1: brute-force 3-NN per query with pos staged in LDS (48 KB of the
// 320 KB WGP pool). One thread per query point; sorted top-3 in registers.
// Outputs normalized inverse-distance weights (PyG clamps d2 at 1e-16).
// ---------------------------------------------------------------------------
__global__ __launch_bounds__(256) void knn_topk_kernel(
    const float* __restrict__ pos,       // [N,3]
    const float* __restrict__ pos_skip,  // [M,3]
    int*   __restrict__ knn_idx,         // [M,3]
    float* __restrict__ knn_w)           // [M,3] (already normalized)
{
    __shared__ float sp[N_PTS * 3];
    const int tid = threadIdx.x;
    for (int i = tid; i < N_PTS * 3; i += 256) sp[i] = pos[i];
    __syncthreads();

    const int m = blockIdx.x * 256 + tid;
    const float qx = pos_skip[3 * m + 0];
    const float qy = pos_skip[3 * m + 1];
    const float qz = pos_skip[3 * m + 2];

    float d0 = 3.4e38f, d1 = 3.4e38f, d2 = 3.4e38f;
    int   i0 = 0, i1 = 0, i2 = 0;

    for (int n = 0; n < N_PTS; ++n) {
        const float dx = qx - sp[3 * n + 0];
        const float dy = qy - sp[3 * n + 1];
        const float dz = qz - sp[3 * n + 2];
        const float d  = dx * dx + dy * dy + dz * dz;
        if (d < d2) {                       // sorted insertion into top-3
            d2 = d; i2 = n;
            if (d2 < d1) {
                float t = d1; d1 = d2; d2 = t;
                int  ti = i1; i1 = i2; i2 = ti;
                if (d1 < d0) {
                    t = d0; d0 = d1; d1 = t;
                    ti = i0; i0 = i1; i1 = ti;
                }
            }
        }
    }

    const float w0 = 1.0f / fmaxf(d0, 1e-16f);
    const float w1 = 1.0f / fmaxf(d1, 1e-16f);
    const float w2 = 1.0f / fmaxf(d2, 1e-16f);
    const float inv = 1.0f / (w0 + w1 + w2);

    knn_idx[3 * m + 0] = i0;
    knn_idx[3 * m + 1] = i1;
    knn_idx[3 * m + 2] = i2;
    knn_w[3 * m + 0] = w0 * inv;
    knn_w[3 * m + 1] = w1 * inv;
    knn_w[3 * m + 2] = w2 * inv;
}

// ---------------------------------------------------------------------------
// Kernel 2: A[m, 0:256]   = sum_k w_k * x[idx_k, :]   (interpolated features)
//           A[m, 256:384] = x_skip[m, :]
// One block per row; fully coalesced column-wise writes.
// ---------------------------------------------------------------------------
__global__ __launch_bounds__(256) void gather_concat_kernel(
    const float* __restrict__ x,        // [N, 256]
    const float* __restrict__ x_skip,   // [M, 128]
    const int*   __restrict__ knn_idx,  // [M, 3]
    const float* __restrict__ knn_w,    // [M, 3]
    float* __restrict__ A)              // [M, 384]
{
    const int m = blockIdx.x;
    const int t = threadIdx.x;

    const int   i0 = knn_idx[3 * m + 0];
    const int   i1 = knn_idx[3 * m + 1];
    const int   i2 = knn_idx[3 * m + 2];
    const float w0 = knn_w[3 * m + 0];
    const float w1 = knn_w[3 * m + 1];
    const float w2 = knn_w[3 * m + 2];

    A[(size_t)m * C_CAT + t] = w0 * x[(size_t)i0 * C_IN + t]
                             + w1 * x[(size_t)i1 * C_IN + t]
                             + w2 * x[(size_t)i2 * C_IN + t];
    if (t < C_SKIP)
        A[(size_t)m * C_CAT + C_IN + t] = x_skip[(size_t)m * C_SKIP + t];
}

// ---------------------------------------------------------------------------
// Kernels 3/4: fp32 GEMM on the tensor pipe via V_WMMA_F32_16X16X4_F32.
// KDIM / NCOLS are template constants so all strides fold into immediate
// load offsets (kills the 64-bit address-arithmetic chains in the K loop).
// Each wave owns one 16x16 output tile; block = 8 waves -> 64x32 block tile.
// A-fragment per ISA layout: lane(l) holds rows M=l%16, K = 2*(l/16)+{0,1}
// -> one aligned v2f load. B-fragment: N=l%16 across lanes, K = 2*(l/16)+{0,1}.
// D: VGPR v -> row (v + 8*half), col l%16. Grid dims are exact multiples of
// the tile so EXEC is all ones (WMMA requirement).
// ---------------------------------------------------------------------------
template <bool RELU, int KDIM, int NCOLS>
__global__ __launch_bounds__(256) void wmma_gemm_kernel(
    const float* __restrict__ A,     // [Mrows, KDIM] row-major
    const float* __restrict__ B,     // [KDIM, NCOLS] row-major
    const float* __restrict__ bias,  // [NCOLS]
    float* __restrict__ C)           // [Mrows, NCOLS]
{
    const int tid  = threadIdx.x;
    const int wave = tid >> 5;
    const int lane = tid & 31;
    const int half = lane >> 4;   // 0: lanes 0-15, 1: lanes 16-31
    const int lm   = lane & 15;

    const int rowBase = blockIdx.y * 64 + (wave & 3) * 16;
    const int colBase = blockIdx.x * 32 + (wave >> 2) * 16;

    const float* Arow = A + (rowBase + lm) * KDIM + 2 * half;
    const float* Bcol = B + (2 * half) * NCOLS + colBase + lm;

    v8f acc = {};
#pragma unroll 8
    for (int k0 = 0; k0 < KDIM; k0 += 4) {
        // A fragment: K = k0 + 2*half + {0,1} for this lane's 16 rows
        v2f a = *(const v2f*)(Arow + k0);
        // B fragment: same K pair, N = colBase + lm
        v2f b;
        b.x = Bcol[k0 * NCOLS];
        b.y = Bcol[(k0 + 1) * NCOLS];
        acc = __builtin_amdgcn_wmma_f32_16x16x4_f32(
            /*neg_a=*/false, a, /*neg_b=*/false, b,
            /*c_mod=*/(short)0, acc, /*reuse_a=*/false, /*reuse_b=*/false);
    }

    const float bn = bias[colBase + lm];
    float* Cp = C + (rowBase + 8 * half) * NCOLS + colBase + lm;
#pragma unroll
    for (int v = 0; v < 8; ++v) {
        float r = acc[v] + bn;
        if (RELU) r = fmaxf(r, 0.0f);
        Cp[v * NCOLS] = r;
    }
}

// ---------------------------------------------------------------------------
// Launch: 4 kernels on `stream`; scratch carved from d_ws:
//   [0, 192K)      knn_idx  (M*3 int)
//   [192K, 384K)   knn_w    (M*3 float)
//   [384K, +24MB)  A        (M x 384 f32)
//   then           H        (M x 256 f32)
// ---------------------------------------------------------------------------
extern "C" void kernel_launch(void* const* d_in, const int* in_sizes, int n_in,
                              void* d_out, int out_size, void* d_ws, size_t ws_size,
                              hipStream_t stream) {
    const float* x        = (const float*)d_in[0];
    const float* pos      = (const float*)d_in[1];
    const float* x_skip   = (const float*)d_in[2];
    const float* pos_skip = (const float*)d_in[3];
    const float* W1       = (const float*)d_in[4];
    const float* b1       = (const float*)d_in[5];
    const float* W2       = (const float*)d_in[6];
    const float* b2       = (const float*)d_in[7];
    float* out = (float*)d_out;

    char*  ws      = (char*)d_ws;
    int*   knn_idx = (int*)ws;
    float* knn_w   = (float*)(ws + (size_t)M_PTS * 3 * sizeof(int));
    float* Amat    = (float*)(ws + (size_t)M_PTS * 3 * sizeof(int) * 2);
    float* Hmat    = Amat + (size_t)M_PTS * C_CAT;

    knn_topk_kernel<<<M_PTS / 256, 256, 0, stream>>>(pos, pos_skip, knn_idx, knn_w);
    gather_concat_kernel<<<M_PTS, 256, 0, stream>>>(x, x_skip, knn_idx, knn_w, Amat);
    wmma_gemm_kernel<true, C_CAT, HDIM><<<dim3(HDIM / 32, M_PTS / 64), 256, 0, stream>>>(
        Amat, W1, b1, Hmat);
    wmma_gemm_kernel<false, HDIM, HDIM><<<dim3(HDIM / 32, M_PTS / 64), 256, 0, stream>>>(
        Hmat, W2, b2, out);
}